// SWAttention_QK_RPB_53085795778935
// MI455X (gfx1250) — compile-verified
//
#include <hip/hip_runtime.h>
#include <hip/hip_bf16.h>
#include <math.h>

typedef __attribute__((ext_vector_type(2))) float v2f;
typedef __attribute__((ext_vector_type(8))) float v8f;

#define HH    56
#define WW    56
#define NPIX  (HH*WW)      // 3136
#define NH    8
#define HD    32
#define WIN   7
#define LL    49
#define CKV   512          // 2*NH*HD
#define TQ    4            // 4x4 query tile -> 16 queries (WMMA M)
#define NT    14           // 56/4 tiles per side
#define UN    10           // union side = TQ + WIN - 1
#define UNION 100          // 10*10 keys in union
#define UPAD  128          // padded to 8 WMMA N-tiles -> 2 tiles per wave, uniform
#define USTR  129          // LDS row stride (odd -> conflict-free across d)

// ---------------------------------------------------------------------------
// Kernel 1: L2-normalize K half of kv into workspace. One 256-thr block per
// (b,n) pixel; each wave32 handles exactly one head's 32 dims (wave32 shuffle
// reduction). Coalesced 256-float read/write per block.
// ---------------------------------------------------------------------------
__global__ __launch_bounds__(256)
void knorm_kernel(const float* __restrict__ kv, float* __restrict__ knorm) {
  int bn = blockIdx.x;            // b*NPIX + n
  int t  = threadIdx.x;           // h*32 + d, 0..255
  float x = kv[(size_t)bn * CKV + t];
  float s = x * x;
  #pragma unroll
  for (int o = 16; o > 0; o >>= 1) s += __shfl_xor(s, o, 32);
  float inv = 1.0f / fmaxf(sqrtf(s), 1e-12f);
  knorm[(size_t)bn * (NH*HD) + t] = x * inv;
}

// ---------------------------------------------------------------------------
// Kernel 2: one workgroup (128 thr = 4 wave32) per (b, h, 4x4 query tile).
//   scores(16 x 128) = Q(16 x 32) x K_union(32 x 128) via v_wmma_f32_16x16x4_f32
// Each wave computes exactly 2 of the 8 N-tiles (uniform trip count -> EXEC
// stays all-1s around WMMA, as the ISA requires). Then gather attn
// (+rpb, -inf where window OOB == reference padding_mask) and v_loc from LDS,
// streamed out with non-temporal stores (outputs > L2, never re-read).
// ---------------------------------------------------------------------------
__global__ __launch_bounds__(128)
void swattn_kernel(const float* __restrict__ q, const float* __restrict__ knorm,
                   const float* __restrict__ kv, const float* __restrict__ rpb,
                   float* __restrict__ attn, float* __restrict__ vloc) {
  __shared__ float qs[16][33];        // Q tile, row-padded
  __shared__ float ks[HD][USTR];      // K union, [d][u]
  __shared__ float vs[HD][USTR];      // V union, [d][u]
  __shared__ float sc[16][UPAD];      // scores

  const int blk  = blockIdx.x;
  const int tile = blk % (NT*NT);
  const int h    = (blk / (NT*NT)) % NH;
  const int b    = blk / (NT*NT*NH);
  const int r0   = (tile / NT) * TQ;
  const int c0   = (tile % NT) * TQ;
  const int tid  = threadIdx.x;
  const int bh   = b*NH + h;

  // ---- load Q tile (16 queries x 32 dims), one b128 per thread ----
  {
    int qi = tid >> 3, d4 = (tid & 7) << 2;
    int n = (r0 + (qi >> 2)) * WW + c0 + (qi & 3);
    float4 qv = *reinterpret_cast<const float4*>(
        q + ((size_t)bh*NPIX + n)*HD + d4);
    qs[qi][d4+0] = qv.x; qs[qi][d4+1] = qv.y;
    qs[qi][d4+2] = qv.z; qs[qi][d4+3] = qv.w;
  }
  // ---- load K/V union (10x10 px x 32 dims) as b128; zero OOB + pad cols ----
  for (int e = tid; e < UPAD*HD/4; e += 128) {
    int u = e >> 3, d4 = (e & 7) << 2;
    float4 kx = make_float4(0.f, 0.f, 0.f, 0.f);
    float4 vx = kx;
    if (u < UNION) {
      int gr = r0 + u/UN - 3, gc = c0 + u%UN - 3;
      if (gr >= 0 && gr < HH && gc >= 0 && gc < WW) {
        size_t nn = (size_t)b*NPIX + gr*WW + gc;
        kx = *reinterpret_cast<const float4*>(knorm + nn*(NH*HD) + h*HD + d4);
        vx = *reinterpret_cast<const float4*>(kv + nn*CKV + NH*HD + h*HD + d4);
      }
    }
    ks[d4+0][u] = kx.x; ks[d4+1][u] = kx.y; ks[d4+2][u] = kx.z; ks[d4+3][u] = kx.w;
    vs[d4+0][u] = vx.x; vs[d4+1][u] = vx.y; vs[d4+2][u] = vx.z; vs[d4+3][u] = vx.w;
  }
  __syncthreads();

  // ---- WMMA GEMM: scores = Q x KU, K=32 as 8 steps of 16x16x4 f32 ----
  const int lane = tid & 31, wave = tid >> 5;
  const int half = lane >> 4, lm = lane & 15;
  #pragma unroll
  for (int tt = 0; tt < 2; ++tt) {                // uniform trip count: EXEC==all-1s
    const int t = wave * 2 + tt;
    v8f acc = {};
    #pragma unroll
    for (int kk = 0; kk < 8; ++kk) {
      // A 16x4 f32 layout: VGPR v, lane L: A[M=L&15, K=kk*4 + 2*(L>>4) + v]
      int k0 = kk*4 + half*2;
      v2f a, bm;
      a.x  = qs[lm][k0];
      a.y  = qs[lm][k0 + 1];
      // B 4x16 f32 (A-symmetric): B[K=k0+v, N=lane&15]
      bm.x = ks[k0][t*16 + lm];
      bm.y = ks[k0 + 1][t*16 + lm];
      acc = __builtin_amdgcn_wmma_f32_16x16x4_f32(
          /*neg_a=*/false, a, /*neg_b=*/false, bm,
          /*c_mod=*/(short)0, acc, /*reuse_a=*/false, /*reuse_b=*/false);
    }
    // D 16x16 f32 layout: VGPR v, lane L: D[M = v + 8*(L>>4), N = L&15]
    #pragma unroll
    for (int v = 0; v < 8; ++v) sc[v + 8*half][t*16 + lm] = acc[v];
  }
  __syncthreads();

  // ---- attn: bias + (-inf where OOB == reference padding_mask) ----
  for (int e = tid; e < 16*LL; e += 128) {
    int qi = e / LL, l = e % LL;
    int dy = l / WIN, dx = l % WIN;
    int qr = qi >> 2, qc = qi & 3;
    int gr = r0 + qr + dy - 3, gc = c0 + qc + dx - 3;
    int n  = (r0 + qr)*WW + c0 + qc;
    float val;
    if (gr >= 0 && gr < HH && gc >= 0 && gc < WW)
      val = sc[qi][(qr + dy)*UN + (qc + dx)] + rpb[h*LL + l];
    else
      val = -INFINITY;
    __builtin_nontemporal_store(val, attn + ((size_t)bh*NPIX + n)*LL + l);
  }

  // ---- v_loc: (b,h,n,d,l) gather; contiguous (d,l) block per query pixel ----
  for (int e = tid; e < 16*HD*LL; e += 128) {
    int qi  = e / (HD*LL);
    int off = e % (HD*LL);
    int d = off / LL, l = off % LL;
    int dy = l / WIN, dx = l % WIN;
    int qr = qi >> 2, qc = qi & 3;
    int gr = r0 + qr + dy - 3, gc = c0 + qc + dx - 3;
    int n  = (r0 + qr)*WW + c0 + qc;
    float val = 0.f;
    if (gr >= 0 && gr < HH && gc >= 0 && gc < WW)
      val = vs[d][(qr + dy)*UN + (qc + dx)];
    __builtin_nontemporal_store(
        val, vloc + ((size_t)bh*NPIX + n)*(size_t)(HD*LL) + off);
  }
}

extern "C" void kernel_launch(void* const* d_in, const int* in_sizes, int n_in,
                              void* d_out, int out_size, void* d_ws, size_t ws_size,
                              hipStream_t stream) {
  const float* kv  = (const float*)d_in[0];   // (B, N, 512)
  const float* q   = (const float*)d_in[1];   // (B, 8, N, 32) pre-normalized
  const float* rpb = (const float*)d_in[2];   // (8, 49)
  // d_in[3] padding_mask unused: mask == window-OOB, recomputed in-kernel.

  const int B = in_sizes[0] / (NPIX * CKV);   // 2

  float* knorm = (float*)d_ws;                            // (B, N, 256) = 6.4 MB
  float* attn  = (float*)d_out;                           // (B, 8, N, 49)
  float* vloc  = attn + (size_t)B*NH*NPIX*LL;             // (B, 8, N, 32, 49)

  knorm_kernel<<<B*NPIX, 256, 0, stream>>>(kv, knorm);
  swattn_kernel<<<B*NH*NT*NT, 128, 0, stream>>>(q, knorm, kv, rpb, attn, vloc);
}